// GraphDecoder_30047591203218
// MI455X (gfx1250) — compile-verified
//
#include <hip/hip_runtime.h>
#include <hip/hip_bf16.h>

typedef __attribute__((ext_vector_type(2))) float v2f;
typedef __attribute__((ext_vector_type(8))) float v8f;

#define WMMA_F32_16x16x4(a, b, c) \
    __builtin_amdgcn_wmma_f32_16x16x4_f32(false, (a), false, (b), (short)0, (c), false, false)

// Problem constants (from reference)
constexpr int Nn   = 50000;   // nodes
constexpr int Mlat = 64;      // latent dim
constexpr int Hh   = 128;     // hidden dim
constexpr int Rr   = 4;       // rounds
constexpr int Ff   = 7;       // output features
constexpr int Ee   = 800000;  // edges
constexpr int Bb   = 32;      // dense-batch graphs
constexpr int NMAX = 512;     // padded nodes per graph

// ---------------------------------------------------------------------------
// Kernel 1: edge_logits[b][n][k] = sum_m zd[b,n,m]*zd[b,k,m] + bias
// One wave -> one 16x16 output tile, 16 x V_WMMA_F32_16X16X4_F32 over M=64.
// Both A and B fragments are contiguous float2 loads from z_dense rows.
// ---------------------------------------------------------------------------
__global__ __launch_bounds__(256) void edge_logits_kernel(
    const float* __restrict__ zd, const float* __restrict__ bias,
    float* __restrict__ out)
{
    const int lane = threadIdx.x & 31;
    const int wave = threadIdx.x >> 5;
    const int tile = blockIdx.x * 8 + wave;     // Bb*32*32 = 32768 tiles total
    const int b  = tile >> 10;                  // tile / 1024
    const int tn = (tile >> 5) & 31;            // n tile
    const int tk = tile & 31;                   // k tile
    const int half = lane >> 4;                 // 0: lanes 0-15, 1: lanes 16-31
    const int r    = lane & 15;

    const float* za = zd + ((size_t)b * NMAX + (size_t)tn * 16 + r) * Mlat;
    const float* zb = zd + ((size_t)b * NMAX + (size_t)tk * 16 + r) * Mlat;

    v8f acc = {};
    #pragma unroll
    for (int k0 = 0; k0 < Mlat; k0 += 4) {
        const int kd = k0 + 2 * half;
        v2f a  = *(const v2f*)(za + kd);  // A[m=r][kd..kd+1]
        v2f bb = *(const v2f*)(zb + kd);  // B[kd..kd+1][n=r]
        acc = WMMA_F32_16x16x4(a, bb, acc);
    }

    const float bv = bias[0];
    #pragma unroll
    for (int v = 0; v < 8; ++v) {
        const int row = tn * 16 + v + 8 * half;
        const int col = tk * 16 + r;
        out[((size_t)b * NMAX + row) * NMAX + col] = acc[v] + bv;
    }
}

// ---------------------------------------------------------------------------
// Kernel 2 (templated): OUT[row][col] = relu(A[row,:K] @ W[:K,col] + bvec[col])
//                       (+ resid[row][col] added after relu when RESID)
// rows = Nn (3125 blocks of one 16-row stripe), cols = Hh (8 waves x 16).
// K and LDA compile-time so the WMMA loop fully unrolls.
// ---------------------------------------------------------------------------
template <int K, int LDA, bool RESID>
__global__ __launch_bounds__(256) void node_gemm_relu_kernel(
    const float* __restrict__ A, const float* __restrict__ W,
    const float* __restrict__ bvec, const float* __restrict__ resid,
    float* __restrict__ out)
{
    const int lane = threadIdx.x & 31;
    const int wave = threadIdx.x >> 5;
    const int half = lane >> 4;
    const int r    = lane & 15;
    const int row0 = blockIdx.x * 16;
    const int col0 = wave * 16;

    const float* arow = A + (size_t)(row0 + r) * LDA;

    v8f acc = {};
    #pragma unroll
    for (int k0 = 0; k0 < K; k0 += 4) {
        const int kd = k0 + 2 * half;
        v2f a = *(const v2f*)(arow + kd);          // A[row=r][kd..kd+1]
        v2f bb;
        bb.x = W[(size_t)kd       * Hh + col0 + r]; // W[kd  ][col=r]
        bb.y = W[(size_t)(kd + 1) * Hh + col0 + r]; // W[kd+1][col=r]
        acc = WMMA_F32_16x16x4(a, bb, acc);
    }

    const float bval = bvec[col0 + r];
    #pragma unroll
    for (int v = 0; v < 8; ++v) {
        const int row = row0 + v + 8 * half;
        const int col = col0 + r;
        float val = fmaxf(acc[v] + bval, 0.0f);
        if (RESID) val += resid[(size_t)row * Hh + col];
        out[(size_t)row * Hh + col] = val;
    }
}

// ---------------------------------------------------------------------------
// Kernel 3: scatter-add.  aggregated[dst[e]] += message[src[e]]  (H=128)
// One wave per edge: lane gathers float4 of message row, 4 fp32 atomics.
// ---------------------------------------------------------------------------
__global__ __launch_bounds__(256) void scatter_kernel(
    const int* __restrict__ ei, const float* __restrict__ msg,
    float* __restrict__ agg)
{
    const int wave = (int)((blockIdx.x * blockDim.x + threadIdx.x) >> 5); // edge id
    const int lane = threadIdx.x & 31;
    if (wave >= Ee) return;

    const int src = ei[wave];       // edge_index[0][e]
    const int dst = ei[Ee + wave];  // edge_index[1][e]

    const float4 m4 = *(const float4*)(msg + (size_t)src * Hh + lane * 4);
    float* a = agg + (size_t)dst * Hh + lane * 4;
    __hip_atomic_fetch_add(a + 0, m4.x, __ATOMIC_RELAXED, __HIP_MEMORY_SCOPE_AGENT);
    __hip_atomic_fetch_add(a + 1, m4.y, __ATOMIC_RELAXED, __HIP_MEMORY_SCOPE_AGENT);
    __hip_atomic_fetch_add(a + 2, m4.z, __ATOMIC_RELAXED, __HIP_MEMORY_SCOPE_AGENT);
    __hip_atomic_fetch_add(a + 3, m4.w, __ATOMIC_RELAXED, __HIP_MEMORY_SCOPE_AGENT);
}

// ---------------------------------------------------------------------------
// Kernel 4: node_logits = state @ W_out + b_out   (N=7, tiny -> plain VALU)
// ---------------------------------------------------------------------------
__global__ __launch_bounds__(256) void out_proj_kernel(
    const float* __restrict__ state, const float* __restrict__ W,
    const float* __restrict__ bo, float* __restrict__ out)
{
    const int gid = blockIdx.x * blockDim.x + threadIdx.x;
    if (gid >= Nn * Ff) return;
    const int node = gid / Ff;
    const int f    = gid % Ff;
    const float* s = state + (size_t)node * Hh;
    float acc = bo[f];
    #pragma unroll 8
    for (int k = 0; k < Hh; ++k) acc += s[k] * W[k * Ff + f];
    out[gid] = acc;
}

// ---------------------------------------------------------------------------
// Host launcher
// ---------------------------------------------------------------------------
extern "C" void kernel_launch(void* const* d_in, const int* in_sizes, int n_in,
                              void* d_out, int out_size, void* d_ws, size_t ws_size,
                              hipStream_t stream) {
    const float* z     = (const float*)d_in[0];   // (N, M)
    const int*   ei    = (const int*)  d_in[1];   // (2, E)
    const float* zd    = (const float*)d_in[2];   // (B, NMAX, M)
    const float* bias  = (const float*)d_in[3];   // (1,)
    const float* W_in  = (const float*)d_in[4];   // (M, H)
    const float* b_in  = (const float*)d_in[5];   // (H,)
    const float* W_msg = (const float*)d_in[6];   // (R, H, H)
    const float* b_msg = (const float*)d_in[7];   // (R, H)
    const float* W_upd = (const float*)d_in[8];   // (R, H, H)
    const float* b_upd = (const float*)d_in[9];   // (R, H)
    const float* W_out = (const float*)d_in[10];  // (H, F)
    const float* b_out = (const float*)d_in[11];  // (F,)

    float* out      = (float*)d_out;
    float* edge_out = out;                                   // B*NMAX*NMAX
    float* node_out = out + (size_t)Bb * NMAX * NMAX;        // N*F

    float* state   = (float*)d_ws;                           // N*H
    float* message = state   + (size_t)Nn * Hh;              // N*H
    float* agg     = message + (size_t)Nn * Hh;              // N*H

    // 1) edge logits: 32768 16x16 tiles, 8 waves/block
    edge_logits_kernel<<<(Bb * 32 * 32) / 8, 256, 0, stream>>>(zd, bias, edge_out);

    // 2) state = relu(z @ W_in + b_in)
    node_gemm_relu_kernel<Mlat, Mlat, false><<<Nn / 16, 256, 0, stream>>>(
        z, W_in, b_in, nullptr, state);

    // 3) message-passing rounds
    for (int rr = 0; rr < Rr; ++rr) {
        node_gemm_relu_kernel<Hh, Hh, false><<<Nn / 16, 256, 0, stream>>>(
            state, W_msg + (size_t)rr * Hh * Hh, b_msg + (size_t)rr * Hh,
            nullptr, message);

        hipMemsetAsync(agg, 0, (size_t)Nn * Hh * sizeof(float), stream);

        scatter_kernel<<<Ee / 8, 256, 0, stream>>>(ei, message, agg);

        // state = state + relu(agg @ W_upd + b_upd)   (element-wise in-place safe)
        node_gemm_relu_kernel<Hh, Hh, true><<<Nn / 16, 256, 0, stream>>>(
            agg, W_upd + (size_t)rr * Hh * Hh, b_upd + (size_t)rr * Hh,
            state, state);
    }

    // 4) node logits
    out_proj_kernel<<<(Nn * Ff + 255) / 256, 256, 0, stream>>>(
        state, W_out, b_out, node_out);
}